// STN_17566416240999
// MI455X (gfx1250) — compile-verified
//
#include <hip/hip_runtime.h>

#define H_OUT 32
#define W_OUT 32
#define HX    128
#define WX    128
#define CCH   32
#define BATCH 32
#define BPQ   64   // 8*8 objects per batch

typedef __attribute__((ext_vector_type(2))) float v2f;
typedef __attribute__((ext_vector_type(4))) float v4f;
typedef __attribute__((ext_vector_type(8))) float v8f;

// ---------------------------------------------------------------------------
// Kernel 1: per-object params (sigmoid/tanh once per object -> table in ws)
// and bbox output computed as a real matmul on V_WMMA_F32_16X16X4_F32:
//   D(16x16) = A(16x4)[sx,sy,tx,ty per entry] x Bcoef(4x16) + C(=0.5 bias)
//   bbox0 = -0.5*sy + 0.5*ty + 0.5
//   bbox1 = -0.5*sx + 0.5*tx + 0.5
//   bbox2 =  0.5*sy + 0.5*ty + 0.5
//   bbox3 =  0.5*sx + 0.5*tx + 0.5
// 2048 entries -> 128 waves -> grid 16 x block 256.
// ---------------------------------------------------------------------------
__global__ __launch_bounds__(256) void stn_params_bbox(
    const float* __restrict__ zw, v4f* __restrict__ tab,
    float* __restrict__ mask)
{
  const int tid    = blockIdx.x * 256 + threadIdx.x;
  const int lane   = tid & 31;
  const int waveId = tid >> 5;          // 0..127
  const int n      = lane & 15;         // N column (A: entry index)
  const int half   = lane >> 4;         // 0: K0/K1 (sx,sy), 1: K2/K3 (tx,ty)
  const int e      = waveId * 16 + n;   // object entry 0..2047

  const v4f z = ((const v4f*)zw)[e];
  const int p  = e & 63;
  const int ii = p >> 3;
  const int jj = p & 7;
  // cr = 0.5 -> bias = 1.5*idx/7 - 0.75
  const float sx = 0.5f / (1.0f + expf(-z.x));
  const float sy = 0.5f / (1.0f + expf(-z.y));
  const float tx = 0.5f * tanhf(z.z) + ((1.5f / 7.0f) * (float)jj - 0.75f);
  const float ty = 0.5f * tanhf(z.w) + ((1.5f / 7.0f) * (float)ii - 0.75f);

#if defined(__HIP_DEVICE_COMPILE__) && __has_builtin(__builtin_amdgcn_wmma_f32_16x16x4_f32)
  // A matrix 16x4 f32: VGPR0 = K0 (lanes 0-15) | K2 (lanes 16-31),
  //                    VGPR1 = K1 (lanes 0-15) | K3 (lanes 16-31)
  v2f a;
  a[0] = half ? tx : sx;
  a[1] = half ? ty : sy;
  // B matrix 4x16 f32 (rows striped across lanes): lanes 0-15 hold rows K0/K1,
  // lanes 16-31 hold rows K2/K3; column = n.
  v2f bm;
  bm[0] = (half == 0) ? ((n == 1) ? -0.5f : ((n == 3) ? 0.5f : 0.0f))
                      : ((n == 1 || n == 3) ? 0.5f : 0.0f);
  bm[1] = (half == 0) ? ((n == 0) ? -0.5f : ((n == 2) ? 0.5f : 0.0f))
                      : ((n == 0 || n == 2) ? 0.5f : 0.0f);
  // C accumulator carries the +0.5 bias in the 4 used columns.
  const float bias = (n < 4) ? 0.5f : 0.0f;
  v8f c;
  #pragma unroll
  for (int v = 0; v < 8; v++) c[v] = bias;

  v8f d = __builtin_amdgcn_wmma_f32_16x16x4_f32(
      /*neg_a=*/false, a, /*neg_b=*/false, bm,
      /*c_mod=*/(short)0, c, /*reuse_a=*/false, /*reuse_b=*/false);

  // D layout: VGPR v, lane l -> M = v + 8*(l>=16), N = l%16.
  // bbox[entry = waveBase + M][N] for N < 4.
  if (n < 4) {
    const int mbase = (waveId * 16 + 8 * half) * 4 + n;
    #pragma unroll
    for (int v = 0; v < 8; v++) mask[mbase + v * 4] = d[v];
  }
#else
  if (half == 0) {
    mask[e * 4 + 0] = 0.5f * ty - 0.5f * sy + 0.5f;
    mask[e * 4 + 1] = 0.5f * tx - 0.5f * sx + 0.5f;
    mask[e * 4 + 2] = 0.5f * ty + 0.5f * sy + 0.5f;
    mask[e * 4 + 3] = 0.5f * tx + 0.5f * sx + 0.5f;
  }
#endif
  if (half == 0) {
    v4f t; t.x = sx; t.y = sy; t.z = tx; t.w = ty;
    tab[e] = t;
  }
}

// ---------------------------------------------------------------------------
// Kernel 2: fused affine-grid + bilinear sample. One block = one (b,p,h) row:
// 32 w-positions x 32 channels = 1024 floats out. Thread = float4 of channels,
// so each corner gather and the store are fully coalesced b128 accesses.
// py and the y-weights are block-uniform. Output is stored NON-TEMPORAL so the
// 268 MB write stream does not evict x (67 MB, ~16x gather reuse) from the
// 192 MB L2 — gathers then stay on-chip and HBM sees ~stores only.
// ---------------------------------------------------------------------------
__global__ __launch_bounds__(256) void stn_sample(
    const float* __restrict__ x, const v4f* __restrict__ tab,
    float* __restrict__ out)
{
  const int row = blockIdx.x;        // ((b*64 + p)*32 + h)
  const int h   = row & 31;
  const int e   = row >> 5;          // b*64 + p
  const int b   = e >> 6;

  const v4f t = tab[e];              // {sx, sy, tx, ty}

  // Block-uniform vertical coordinate.
  const float Y   = (2.0f / 31.0f) * (float)h - 1.0f;
  const float py  = 63.5f * (t.y * Y + t.w) + 63.5f;
  const float y0f = floorf(py);
  const float y0c = fminf(fmaxf(y0f, 0.0f), 127.0f);
  const float y1c = fminf(fmaxf(y0f + 1.0f, 0.0f), 127.0f);
  const float wya = y1c - py;        // weight for y0 row
  const float wyb = py - y0c;        // weight for y1 row
  const int   yi0 = (int)y0c;
  const int   yi1 = (int)y1c;

  const int wi = threadIdx.x >> 3;   // 0..31 output column
  const int c4 = threadIdx.x & 7;    // 0..7  channel vector

  const float X   = (2.0f / 31.0f) * (float)wi - 1.0f;
  const float px  = 63.5f * (t.x * X + t.z) + 63.5f;
  const float x0f = floorf(px);
  const float x0c = fminf(fmaxf(x0f, 0.0f), 127.0f);
  const float x1c = fminf(fmaxf(x0f + 1.0f, 0.0f), 127.0f);
  const float wxa = x1c - px;
  const float wxc = px - x0c;
  const int   xi0 = (int)x0c;
  const int   xi1 = (int)x1c;

  const float wa = wxa * wya;
  const float wb = wxa * wyb;
  const float wc = wxc * wya;
  const float wd = wxc * wyb;

  const float* xb = x + ((size_t)b * (HX * WX * CCH));
  const int cO = c4 << 2;
  const v4f Ia = *(const v4f*)(xb + ((yi0 * WX + xi0) * CCH + cO));
  const v4f Ib = *(const v4f*)(xb + ((yi1 * WX + xi0) * CCH + cO));
  const v4f Ic = *(const v4f*)(xb + ((yi0 * WX + xi1) * CCH + cO));
  const v4f Id = *(const v4f*)(xb + ((yi1 * WX + xi1) * CCH + cO));

  const v4f o = wa * Ia + wb * Ib + wc * Ic + wd * Id;

  // Streaming (non-temporal) 16B store: keep L2 for the gathered image.
  __builtin_nontemporal_store(
      o, (v4f*)(out + ((size_t)row * (W_OUT * CCH) + (wi * CCH + cO))));
}

extern "C" void kernel_launch(void* const* d_in, const int* in_sizes, int n_in,
                              void* d_out, int out_size, void* d_ws, size_t ws_size,
                              hipStream_t stream) {
  const float* x  = (const float*)d_in[0];   // (32,128,128,32) f32
  const float* zw = (const float*)d_in[1];   // (32,8,8,4) f32

  float* out  = (float*)d_out;                                   // 67,108,864
  float* mask = out + (size_t)BATCH * BPQ * H_OUT * W_OUT * CCH; // + 8,192

  v4f* tab = (v4f*)d_ws;   // 2048 * 16 B = 32 KB scratch

  stn_params_bbox<<<16, 256, 0, stream>>>(zw, tab, mask);
  stn_sample<<<BATCH * BPQ * H_OUT, 256, 0, stream>>>(x, tab, out);
}